// BalanceCrossEntropyLoss_88510686036525
// MI455X (gfx1250) — compile-verified
//
#include <hip/hip_runtime.h>
#include <hip/hip_bf16.h>
#include <cstdint>

#define TPB 256
#define NB  2048

// ---------------------------------------------------------------------------
// Shared state for the radix select, lives in d_ws.
// ---------------------------------------------------------------------------
struct OhemState {
    float    pos_sum;    // sum of losses on positive area
    unsigned pos_count;  // #(positive_area > 0.5)
    unsigned neg_avail;  // #(negative_area > 0.5)
    unsigned k;          // negative_count = min(neg_avail, int(pos*3))
    unsigned k_rem;      // remaining k inside current radix bin
    unsigned prefix;     // accumulated high bits of threshold key
    unsigned T;          // final threshold key (elements with key > T fully in)
    unsigned m;          // #elements equal to T to include
};

// ---------------------------------------------------------------------------
// Elementwise math
// ---------------------------------------------------------------------------
__device__ __forceinline__ float bce_loss(float pv, float yv, float wv) {
    float pc = fminf(fmaxf(pv,        1e-12f), 1.0f);
    float qc = fminf(fmaxf(1.0f - pv, 1e-12f), 1.0f);
    return -wv * (yv * logf(pc) + (1.0f - yv) * logf(qc));
}

// key: monotone map of non-negative float loss; 0 reserved for "not negative"
__device__ __forceinline__ unsigned loss_key(float l) {
    return __float_as_uint(l) + 1u;
}

// ---------------------------------------------------------------------------
// CDNA5 async global->LDS staging (GVS mode: sgpr base + 32-bit vgpr offset)
// ---------------------------------------------------------------------------
__device__ __forceinline__ void async_ld128(const float* base, unsigned byteOff,
                                            unsigned ldsOff) {
    asm volatile("global_load_async_to_lds_b128 %0, %1, %2"
                 :
                 : "v"(ldsOff), "v"(byteOff),
                   "s"((unsigned long long)(uintptr_t)base)
                 : "memory");
}
#define WAIT_ASYNC_LE(n) asm volatile("s_wait_asynccnt " #n ::: "memory")

// ---------------------------------------------------------------------------
// Wave(32) sum via V_WMMA_F32_16X16X4_F32:
//   A[16x4]: lane L<16 -> vgpr0=A[L][0], vgpr1=A[L][1]; lane>=16 -> K=2,3.
//   Put lane value in vgpr0, 0 in vgpr1; B = ones(4x16).
//   D[m][n] = s[m] + s[m+16]  (independent of n).
//   Lane sums its 8 D regs: lane0 holds rows 0..7, lane16 rows 8..15.
// ---------------------------------------------------------------------------
typedef __attribute__((ext_vector_type(2))) float v2f;
typedef __attribute__((ext_vector_type(8))) float v8f;

__device__ __forceinline__ float wave_sum32(float v) {
    v2f a; a.x = v;    a.y = 0.0f;
    v2f b; b.x = 1.0f; b.y = 1.0f;
    v8f c = {};
    c = __builtin_amdgcn_wmma_f32_16x16x4_f32(false, a, false, b,
                                              (short)0, c, false, false);
    float r = c[0] + c[1] + c[2] + c[3] + c[4] + c[5] + c[6] + c[7];
    int ri = __float_as_int(r);
    float lo = __int_as_float(__builtin_amdgcn_readlane(ri, 0));
    float hi = __int_as_float(__builtin_amdgcn_readlane(ri, 16));
    return lo + hi;
}

// ---------------------------------------------------------------------------
// Pass 1: stream all 4 inputs via async-LDS double buffering.
// Per block: partial pos_sum / pos_count / neg_count; global 2048-bin
// histogram of key[31:21] (LDS-accumulated, integer atomics -> deterministic).
// ---------------------------------------------------------------------------
__global__ __launch_bounds__(TPB) void pass_main(
    const float* __restrict__ p, const float* __restrict__ y,
    const float* __restrict__ mk, const float* __restrict__ w,
    unsigned N, float* __restrict__ bps, unsigned* __restrict__ bpc,
    unsigned* __restrict__ bnc, unsigned* __restrict__ hist0)
{
    __shared__ __align__(16) float lbuf[2][4][TPB * 4];  // 32 KB
    __shared__ unsigned shist[2048];                     // 8 KB
    __shared__ float    rs[TPB];
    __shared__ unsigned rc[TPB], rn[TPB];

    const unsigned t = threadIdx.x;
    for (unsigned b = t; b < 2048u; b += TPB) shist[b] = 0u;
    __syncthreads();

    const unsigned N4  = N >> 2;
    const unsigned gsz = gridDim.x * blockDim.x;
    const unsigned s0  = blockIdx.x * blockDim.x + t;

    float pos_sum = 0.0f;
    unsigned posc = 0u, negc = 0u;
    const float* bases[4] = { p, y, mk, w };

    if (N4 > 0u) {
        const unsigned nIter = (N4 + gsz - 1u) / gsz;
        unsigned buf = 0u;
        {   // prologue: stage chunk 0
            unsigned sc   = s0 < N4 ? s0 : (N4 - 1u);
            unsigned boff = sc * 16u;
#pragma unroll
            for (int a = 0; a < 4; ++a)
                async_ld128(bases[a], boff,
                            (unsigned)(uintptr_t)&lbuf[0][a][t * 4]);
        }
        for (unsigned i = 0; i < nIter; ++i) {
            const unsigned sCur = s0 + i * gsz;
            if (i + 1u < nIter) {   // stage next chunk, then wait for current
                unsigned sn   = s0 + (i + 1u) * gsz;
                unsigned sc   = sn < N4 ? sn : (N4 - 1u);
                unsigned boff = sc * 16u;
#pragma unroll
                for (int a = 0; a < 4; ++a)
                    async_ld128(bases[a], boff,
                                (unsigned)(uintptr_t)&lbuf[buf ^ 1u][a][t * 4]);
                WAIT_ASYNC_LE(4);   // 8 outstanding; oldest 4 (current) done
            } else {
                WAIT_ASYNC_LE(0);
            }
            if (sCur < N4) {
                float4 P = *(const float4*)&lbuf[buf][0][t * 4];
                float4 Y = *(const float4*)&lbuf[buf][1][t * 4];
                float4 M = *(const float4*)&lbuf[buf][2][t * 4];
                float4 W = *(const float4*)&lbuf[buf][3][t * 4];
                const float pv[4] = {P.x, P.y, P.z, P.w};
                const float yv[4] = {Y.x, Y.y, Y.z, Y.w};
                const float mv[4] = {M.x, M.y, M.z, M.w};
                const float wv[4] = {W.x, W.y, W.z, W.w};
#pragma unroll
                for (int j = 0; j < 4; ++j) {
                    float l   = bce_loss(pv[j], yv[j], wv[j]);
                    bool  msk = mv[j] > 0.5f;
                    bool  pos = (yv[j] > 0.5f) && msk;
                    bool  neg = (yv[j] < 0.5f) && msk;
                    if (pos) { pos_sum += l; ++posc; }
                    if (neg) {
                        ++negc;
                        atomicAdd(&shist[loss_key(l) >> 21], 1u);
                    }
                }
            }
            buf ^= 1u;
        }
    }
    // scalar tail (N % 4), handled once
    if (blockIdx.x == 0 && t == 0) {
        for (unsigned e = N4 << 2; e < N; ++e) {
            float l   = bce_loss(p[e], y[e], w[e]);
            bool  msk = mk[e] > 0.5f;
            if ((y[e] > 0.5f) && msk) { pos_sum += l; ++posc; }
            if ((y[e] < 0.5f) && msk) {
                ++negc;
                atomicAdd(&shist[loss_key(l) >> 21], 1u);
            }
        }
    }
    // deterministic block tree reduce
    rs[t] = pos_sum; rc[t] = posc; rn[t] = negc;
    __syncthreads();
    for (unsigned o = TPB / 2; o > 0; o >>= 1) {
        if (t < o) { rs[t] += rs[t + o]; rc[t] += rc[t + o]; rn[t] += rn[t + o]; }
        __syncthreads();
    }
    if (t == 0) { bps[blockIdx.x] = rs[0]; bpc[blockIdx.x] = rc[0]; bnc[blockIdx.x] = rn[0]; }
    for (unsigned b = t; b < 2048u; b += TPB) {
        unsigned c = shist[b];
        if (c) atomicAdd(&hist0[b], c);
    }
}

// ---------------------------------------------------------------------------
// Single-wave reduce of block partials (WMMA) + level-0 radix scan.
// ---------------------------------------------------------------------------
__global__ void reduce_select0(const float* __restrict__ bps,
                               const unsigned* __restrict__ bpc,
                               const unsigned* __restrict__ bnc,
                               const unsigned* __restrict__ hist0,
                               OhemState* __restrict__ st, int nb)
{
    const int lane = threadIdx.x;
    float s = 0.0f, cp = 0.0f, cn = 0.0f;   // counts < 2^24 -> exact in f32
    for (int i = lane; i < nb; i += 32) {
        s += bps[i]; cp += (float)bpc[i]; cn += (float)bnc[i];
    }
    float ts  = wave_sum32(s);
    float tcp = wave_sum32(cp);
    float tcn = wave_sum32(cn);
    if (lane == 0) {
        unsigned pos_count = (unsigned)(tcp + 0.5f);
        unsigned neg_avail = (unsigned)(tcn + 0.5f);
        int k3 = (int)((float)pos_count * 3.0f);   // matches astype(int32)
        if (k3 < 0) k3 = 0;
        unsigned k = (unsigned)k3;
        if (k > neg_avail) k = neg_avail;
        st->pos_sum = ts; st->pos_count = pos_count; st->neg_avail = neg_avail;
        st->k = k; st->prefix = 0u; st->k_rem = k; st->T = 0u; st->m = 0u;
        unsigned above = 0u;
        for (int b = 2047; b >= 0; --b) {
            unsigned c = hist0[b];
            if (above + c >= k) { st->prefix = (unsigned)b; st->k_rem = k - above; break; }
            above += c;
        }
    }
}

// ---------------------------------------------------------------------------
// Histogram refinement pass (generic level): keys matching current prefix.
// ---------------------------------------------------------------------------
__global__ __launch_bounds__(TPB) void pass_hist(
    const float* __restrict__ p, const float* __restrict__ y,
    const float* __restrict__ mk, const float* __restrict__ w,
    unsigned N, const OhemState* __restrict__ st, unsigned* __restrict__ ghist,
    int cmpShift, int binShift, unsigned binMask, int nbins)
{
    __shared__ unsigned sh[2048];
    for (int b = threadIdx.x; b < nbins; b += TPB) sh[b] = 0u;
    __syncthreads();
    const unsigned prefix = st->prefix;
    const unsigned N4 = N >> 2, gsz = gridDim.x * blockDim.x;
    const float4* p4 = (const float4*)p;
    const float4* y4 = (const float4*)y;
    const float4* m4 = (const float4*)mk;
    const float4* w4 = (const float4*)w;
    for (unsigned s = blockIdx.x * blockDim.x + threadIdx.x; s < N4; s += gsz) {
        unsigned pf = (s + gsz < N4) ? s + gsz : s;
        __builtin_prefetch(&p4[pf], 0, 1);
        __builtin_prefetch(&w4[pf], 0, 1);
        float4 P = p4[s], Y = y4[s], M = m4[s], W = w4[s];
        const float pv[4] = {P.x, P.y, P.z, P.w};
        const float yv[4] = {Y.x, Y.y, Y.z, Y.w};
        const float mv[4] = {M.x, M.y, M.z, M.w};
        const float wv[4] = {W.x, W.y, W.z, W.w};
#pragma unroll
        for (int j = 0; j < 4; ++j) {
            if ((yv[j] < 0.5f) && (mv[j] > 0.5f)) {
                unsigned key = loss_key(bce_loss(pv[j], yv[j], wv[j]));
                if ((key >> cmpShift) == prefix)
                    atomicAdd(&sh[(key >> binShift) & binMask], 1u);
            }
        }
    }
    if (blockIdx.x == 0 && threadIdx.x == 0) {
        for (unsigned e = N4 << 2; e < N; ++e) {
            if ((y[e] < 0.5f) && (mk[e] > 0.5f)) {
                unsigned key = loss_key(bce_loss(p[e], y[e], w[e]));
                if ((key >> cmpShift) == prefix)
                    atomicAdd(&ghist[(key >> binShift) & binMask], 1u);
            }
        }
    }
    __syncthreads();
    for (int b = threadIdx.x; b < nbins; b += TPB)
        if (sh[b]) atomicAdd(&ghist[b], sh[b]);
}

// ---------------------------------------------------------------------------
// Descending scan of one refinement level; final level fixes T and m.
// ---------------------------------------------------------------------------
__global__ void select_level(const unsigned* __restrict__ h, int nbins,
                             int shift, int isFinal, OhemState* __restrict__ st)
{
    if (threadIdx.x == 0 && blockIdx.x == 0) {
        const unsigned k = st->k_rem;
        unsigned above = 0u;
        for (int b = nbins - 1; b >= 0; --b) {
            unsigned c = h[b];
            if (above + c >= k) {
                st->prefix = (st->prefix << shift) | (unsigned)b;
                st->k_rem  = k - above;
                if (isFinal) { st->T = st->prefix; st->m = st->k_rem; }
                return;
            }
            above += c;
        }
        st->prefix <<= shift; st->k_rem = 0u;        // unreachable fallback
        if (isFinal) { st->T = 0u; st->m = 0u; }
    }
}

// ---------------------------------------------------------------------------
// Final pass: per-block partial sum of negative losses with key > T.
// ---------------------------------------------------------------------------
__global__ __launch_bounds__(TPB) void pass_final(
    const float* __restrict__ p, const float* __restrict__ y,
    const float* __restrict__ mk, const float* __restrict__ w,
    unsigned N, const OhemState* __restrict__ st, float* __restrict__ sab)
{
    __shared__ float rs[TPB];
    const unsigned T = st->T;
    const unsigned N4 = N >> 2, gsz = gridDim.x * blockDim.x;
    const float4* p4 = (const float4*)p;
    const float4* y4 = (const float4*)y;
    const float4* m4 = (const float4*)mk;
    const float4* w4 = (const float4*)w;
    float acc = 0.0f;
    for (unsigned s = blockIdx.x * blockDim.x + threadIdx.x; s < N4; s += gsz) {
        unsigned pf = (s + gsz < N4) ? s + gsz : s;
        __builtin_prefetch(&p4[pf], 0, 1);
        float4 P = p4[s], Y = y4[s], M = m4[s], W = w4[s];
        const float pv[4] = {P.x, P.y, P.z, P.w};
        const float yv[4] = {Y.x, Y.y, Y.z, Y.w};
        const float mv[4] = {M.x, M.y, M.z, M.w};
        const float wv[4] = {W.x, W.y, W.z, W.w};
#pragma unroll
        for (int j = 0; j < 4; ++j) {
            if ((yv[j] < 0.5f) && (mv[j] > 0.5f)) {
                float l = bce_loss(pv[j], yv[j], wv[j]);
                if (loss_key(l) > T) acc += l;
            }
        }
    }
    if (blockIdx.x == 0 && threadIdx.x == 0) {
        for (unsigned e = N4 << 2; e < N; ++e) {
            if ((y[e] < 0.5f) && (mk[e] > 0.5f)) {
                float l = bce_loss(p[e], y[e], w[e]);
                if (loss_key(l) > T) acc += l;
            }
        }
    }
    rs[threadIdx.x] = acc;
    __syncthreads();
    for (unsigned o = TPB / 2; o > 0; o >>= 1) {
        if (threadIdx.x < o) rs[threadIdx.x] += rs[threadIdx.x + o];
        __syncthreads();
    }
    if (threadIdx.x == 0) sab[blockIdx.x] = rs[0];
}

// ---------------------------------------------------------------------------
// Final combine (single wave, WMMA reduce).
// ---------------------------------------------------------------------------
__global__ void reduce_final(const float* __restrict__ sab, int nb,
                             const OhemState* __restrict__ st,
                             float* __restrict__ out)
{
    const int lane = threadIdx.x;
    float s = 0.0f;
    for (int i = lane; i < nb; i += 32) s += sab[i];
    float S = wave_sum32(s);
    if (lane == 0) {
        float extra = (st->m > 0u)
                          ? (float)st->m * __uint_as_float(st->T - 1u)
                          : 0.0f;
        float denom = (float)(st->pos_count + st->k) + 1e-6f;
        out[0] = (st->pos_sum + S + extra) / denom;
    }
}

// ---------------------------------------------------------------------------
// Host launch.  ws layout (bytes):
//   [0,8192)      hist0 (2048 u32)
//   [8192,16384)  hist1 (2048 u32)
//   [16384,20480) hist2 (1024 u32)
//   [20480,20608) OhemState (+pad)
//   [20608,...)   bps/bpc/bnc/sab, NB entries each
// ---------------------------------------------------------------------------
extern "C" void kernel_launch(void* const* d_in, const int* in_sizes, int n_in,
                              void* d_out, int out_size, void* d_ws,
                              size_t ws_size, hipStream_t stream)
{
    (void)n_in; (void)out_size; (void)ws_size;
    const float* p  = (const float*)d_in[0];
    const float* y  = (const float*)d_in[1];
    const float* mk = (const float*)d_in[2];
    const float* w  = (const float*)d_in[3];
    const unsigned N = (unsigned)in_sizes[0];

    char* ws = (char*)d_ws;
    unsigned*  hist0 = (unsigned*)(ws);
    unsigned*  hist1 = (unsigned*)(ws + 8192);
    unsigned*  hist2 = (unsigned*)(ws + 16384);
    OhemState* st    = (OhemState*)(ws + 20480);
    float*     bps   = (float*)(ws + 20608);
    unsigned*  bpc   = (unsigned*)(ws + 20608 + 4 * NB);
    unsigned*  bnc   = (unsigned*)(ws + 20608 + 8 * NB);
    float*     sab   = (float*)(ws + 20608 + 12 * NB);

    hipMemsetAsync(d_ws, 0, 20608, stream);   // hists + state
    pass_main<<<NB, TPB, 0, stream>>>(p, y, mk, w, N, bps, bpc, bnc, hist0);
    reduce_select0<<<1, 32, 0, stream>>>(bps, bpc, bnc, hist0, st, NB);
    pass_hist<<<NB, TPB, 0, stream>>>(p, y, mk, w, N, st, hist1,
                                      21, 10, 0x7FFu, 2048);
    select_level<<<1, 1, 0, stream>>>(hist1, 2048, 11, 0, st);
    pass_hist<<<NB, TPB, 0, stream>>>(p, y, mk, w, N, st, hist2,
                                      10, 0, 0x3FFu, 1024);
    select_level<<<1, 1, 0, stream>>>(hist2, 1024, 10, 1, st);
    pass_final<<<NB, TPB, 0, stream>>>(p, y, mk, w, N, st, sab);
    reduce_final<<<1, 32, 0, stream>>>(sab, NB, st, (float*)d_out);
}